// EntanglementRegularizer_71631464563318
// MI455X (gfx1250) — compile-verified
//
#include <hip/hip_runtime.h>
#include <hip/hip_bf16.h>

typedef __attribute__((ext_vector_type(2))) float v2f;
typedef __attribute__((ext_vector_type(8))) float v8f;

#define NUM_BINS 256
#define WEIGHT_C 0.01f
#define EPS_C 1e-10f
#define L2E 1.44269504088896340736f

// ---------------------------------------------------------------------------
// Zero the 256-float global histogram accumulator (in d_ws).
// ---------------------------------------------------------------------------
__global__ void zero_hist_kernel(float* __restrict__ g_hist) {
    g_hist[threadIdx.x] = 0.0f;
}

// ---------------------------------------------------------------------------
// Main kernel: soft-histogram via f32 WMMA + v_exp_f32.
// Each wave processes 16 elements per tile-iteration:
//   A (16x4 f32): rows = x-values, K-columns = {x, x^2, 1, 0}
//   B (4x16 f32): per bin-tile t, K-rows = {4*L2E*b, -2*L2E, -2*L2E*b^2, 0}
//   D = A*B = log2(e) * [ -2*(x-b)^2 ]  -> e = exp2(D) = softmax numerator
// ---------------------------------------------------------------------------
__global__ __launch_bounds__(256) void hist_kernel(const float* __restrict__ y,
                                                   const float* __restrict__ bins,
                                                   float* __restrict__ g_hist,
                                                   long long N, int total_waves) {
    __shared__ float lds_hist[NUM_BINS];
    lds_hist[threadIdx.x] = 0.0f;
    __syncthreads();

    const int lane  = threadIdx.x & 31;
    const int col   = lane & 15;
    const bool hi   = lane >= 16;
    const int gwave = blockIdx.x * 8 + (threadIdx.x >> 5);

    // B operands: constant across the whole element loop (32 VGPRs).
    // A-layout mirror: lanes 0-15 hold K=0,1 ; lanes 16-31 hold K=2,3.
    v2f Bt[16];
#pragma unroll
    for (int t = 0; t < 16; ++t) {
        float b = bins[t * 16 + col];
        if (!hi) { Bt[t].x = 4.0f * L2E * b;      Bt[t].y = -2.0f * L2E; }
        else     { Bt[t].x = -2.0f * L2E * b * b; Bt[t].y = 0.0f; }
    }

    float histreg[16];
#pragma unroll
    for (int t = 0; t < 16; ++t) histreg[t] = 0.0f;

    const long long ntiles = (N + 15) >> 4;
    for (long long tile = gwave; tile < ntiles; tile += total_waves) {
        const long long base = tile << 4;

        float x = 0.0f;
        if (!hi) {
            long long idx = base + col;
            if (idx < N) x = y[idx];
        }
        v2f a;
        if (!hi) { a.x = x;    a.y = x * x; }   // K=0: x, K=1: x^2
        else     { a.x = 1.0f; a.y = 0.0f; }    // K=2: 1, K=3: 0

        float est[16][8];   // exp values for all 256 bins of this 16-elem group
        v8f rowsum = {};

#pragma unroll
        for (int t = 0; t < 16; ++t) {
            v8f c = {};
            // 8 args: (neg_a, A, neg_b, B, c_mod, C, reuse_a, reuse_b)
            v8f d = __builtin_amdgcn_wmma_f32_16x16x4_f32(
                false, a, false, Bt[t], (short)0, c, false, false);
#pragma unroll
            for (int r = 0; r < 8; ++r) {
                float e = __builtin_amdgcn_exp2f(d[r]);   // v_exp_f32
                est[t][r] = e;
                rowsum[r] += e;
            }
        }

        // Finish row sums across the 16 lanes of each half, take reciprocal.
        float recip[8];
#pragma unroll
        for (int r = 0; r < 8; ++r) {
            float s = rowsum[r];
            s += __shfl_xor(s, 1, 32);
            s += __shfl_xor(s, 2, 32);
            s += __shfl_xor(s, 4, 32);
            s += __shfl_xor(s, 8, 32);
            long long row = base + r + (hi ? 8 : 0);
            recip[r] = (row < N) ? __builtin_amdgcn_rcpf(s) : 0.0f;
        }

        // hist_j += sum_i e_ij / S_i   (combine the two 8-row halves via xor-16)
#pragma unroll
        for (int t = 0; t < 16; ++t) {
            float p = 0.0f;
#pragma unroll
            for (int r = 0; r < 8; ++r) p += est[t][r] * recip[r];
            p += __shfl_xor(p, 16, 32);
            histreg[t] += p;
        }
    }

    // Block reduction: lanes 0-15 of each wave hold unique column partials
    // (hi half is a duplicate after the xor-16 add).
    if (!hi) {
#pragma unroll
        for (int t = 0; t < 16; ++t)
            atomicAdd(&lds_hist[t * 16 + col], histreg[t]);   // ds_add_f32
    }
    __syncthreads();
    atomicAdd(&g_hist[threadIdx.x], lds_hist[threadIdx.x]);   // global_atomic_add_f32
}

// ---------------------------------------------------------------------------
// Finalize: hist = accum/N; normalize; out = WEIGHT * sum(h*log(h+eps))
// ( == -entropy * WEIGHT, matching the reference exactly )
// ---------------------------------------------------------------------------
__global__ void finalize_kernel(const float* __restrict__ g_hist,
                                float* __restrict__ out, float invN) {
    __shared__ float red[NUM_BINS];
    const int tid = threadIdx.x;
    float v = g_hist[tid] * invN;
    red[tid] = v;
    __syncthreads();
    for (int s = 128; s > 0; s >>= 1) {
        if (tid < s) red[tid] += red[tid + s];
        __syncthreads();
    }
    const float total = red[0];
    __syncthreads();
    const float h = v / (total + EPS_C);
    red[tid] = h * __logf(h + EPS_C);
    __syncthreads();
    for (int s = 128; s > 0; s >>= 1) {
        if (tid < s) red[tid] += red[tid + s];
        __syncthreads();
    }
    if (tid == 0) out[0] = WEIGHT_C * red[0];
}

// ---------------------------------------------------------------------------
extern "C" void kernel_launch(void* const* d_in, const int* in_sizes, int n_in,
                              void* d_out, int out_size, void* d_ws, size_t ws_size,
                              hipStream_t stream) {
    const float* y    = (const float*)d_in[0];
    const float* bins = (const float*)d_in[1];
    float* out    = (float*)d_out;
    float* g_hist = (float*)d_ws;           // 256 floats of scratch

    const long long N = (long long)in_sizes[0];
    const int blocks = 512;                  // 4096 waves; ~32 tile-iters each
    const int total_waves = blocks * 8;

    zero_hist_kernel<<<1, NUM_BINS, 0, stream>>>(g_hist);
    hist_kernel<<<blocks, 256, 0, stream>>>(y, bins, g_hist, N, total_waves);
    finalize_kernel<<<1, NUM_BINS, 0, stream>>>(g_hist, out, 1.0f / (float)N);
}